// BoxBlur2d_75058848465571
// MI455X (gfx1250) — compile-verified
//
#include <hip/hip_runtime.h>
#include <cstdint>

// Depthwise 11x11 box blur, reflect padding, (8,64,512,512) f32.
// Bandwidth-bound: ~1.07 GB HBM traffic -> ~46us floor @ 23.3 TB/s.
// CDNA5 paths: GLOBAL_LOAD_ASYNC_TO_LDS_B128/B32 (ASYNCcnt) for the tile
// stage-in, s_wait_asynccnt fencing, ds_load_b128 LDS vector reads,
// non-temporal b128 streaming stores.

#define KW        11
#define PAD       5                        // KW/2
#define IMG_H     512
#define IMG_W     512
#define TILE_H    32
#define TILE_W    128
#define IN_H      (TILE_H + KW - 1)        // 42 rows (with halo)
#define LOAD_W    144                      // loaded cols: [col0-8, col0+136) -> 36 x 4-float chunks
#define CHUNKS    (LOAD_W / 4)             // 36 b128 chunks per row
#define IN_STRIDE 148                      // floats; 592 B = 37*16 -> rows 16B-aligned
#define HS_STRIDE 132                      // floats; 528 B = 33*16 -> rows 16B-aligned
#define NTHREADS  256

typedef float v4f __attribute__((ext_vector_type(4)));

__device__ __forceinline__ int reflect_idx(int i, int n) {
    // jnp.pad mode='reflect' (mirror, no edge duplication)
    i = (i < 0) ? -i : i;
    i = (i >= n) ? (2 * n - 2 - i) : i;
    return i;
}

__device__ __forceinline__ void async_b128(uint32_t lds_off, uint64_t gaddr) {
    asm volatile("global_load_async_to_lds_b128 %0, %1, off"
                 :: "v"(lds_off), "v"(gaddr) : "memory");
}
__device__ __forceinline__ void async_b32(uint32_t lds_off, uint64_t gaddr) {
    asm volatile("global_load_async_to_lds_b32 %0, %1, off"
                 :: "v"(lds_off), "v"(gaddr) : "memory");
}

__global__ __launch_bounds__(NTHREADS)
void box_blur11_kernel(const float* __restrict__ x, float* __restrict__ y) {
    __shared__ __align__(16) float s_in[IN_H * IN_STRIDE]; // raw padded tile
    __shared__ __align__(16) float s_h [IN_H * HS_STRIDE]; // horizontal sums

    const int tid     = threadIdx.x;
    const int plane   = blockIdx.z;                 // b*64 + c
    const int row0    = blockIdx.y * TILE_H;
    const int col0    = blockIdx.x * TILE_W;
    const int c_start = col0 - 8;                   // 16B-aligned global col of LDS col 0

    const float* __restrict__ xp = x + (size_t)plane * (IMG_H * IMG_W);
    float*       __restrict__ yp = y + (size_t)plane * (IMG_H * IMG_W);

    // ---- Phase 1a: bulk async B128 row-contiguous loads ----
    // y-reflection remaps whole rows (still contiguous); only x-edges need fixup.
    const int total_chunks = IN_H * CHUNKS;                       // 1512
    #pragma unroll 1
    for (int it = 0; it < (total_chunks + NTHREADS - 1) / NTHREADS; ++it) {
        int task = it * NTHREADS + tid;
        if (task >= total_chunks) task = total_chunks - 1;        // benign duplicate
        const int r  = task / CHUNKS;
        const int k  = task - r * CHUNKS;
        const int gc = c_start + 4 * k;
        if (gc >= 0 && gc + 4 <= IMG_W) {                         // skip OOB chunks
            const int gr = reflect_idx(row0 + r - PAD, IMG_H);
            async_b128((uint32_t)(uintptr_t)&s_in[r * IN_STRIDE + 4 * k],
                       (uint64_t)(uintptr_t)(xp + (size_t)gr * IMG_W + gc));
        }
    }

    // ---- Phase 1b: x-border reflected columns (5 per row), scalar async ----
    if (col0 == 0 || col0 + TILE_W == IMG_W) {
        const bool left = (col0 == 0);
        if (tid < IN_H * PAD) {                                   // 210 tasks
            const int lr  = tid / PAD;
            const int j   = tid - lr * PAD;
            const int v   = left ? (j - PAD) : (IMG_W + j);       // virtual padded col
            const int gc  = left ? (-v) : (2 * IMG_W - 2 - v);    // reflected source
            const int lc2 = v - c_start;                          // 3..7 or 136..140
            const int gr  = reflect_idx(row0 + lr - PAD, IMG_H);
            async_b32((uint32_t)(uintptr_t)&s_in[lr * IN_STRIDE + lc2],
                      (uint64_t)(uintptr_t)(xp + (size_t)gr * IMG_W + gc));
        }
    }

    asm volatile("s_wait_asynccnt 0" ::: "memory");
    __syncthreads();

    // ---- Phase 2: horizontal 11-tap sliding sums ----
    // 42 rows x 8 segments of 16 output cols = 336 tasks.
    // First needed input col is (col0-5) -> LDS col 3.
    #pragma unroll 1
    for (int task = tid; task < IN_H * (TILE_W / 16); task += NTHREADS) {
        const int r  = task >> 3;
        const int c0 = (task & 7) * 16;
        const float* __restrict__ rin  = &s_in[r * IN_STRIDE + 3 + c0];
        float*       __restrict__ rout = &s_h [r * HS_STRIDE + c0];
        float sum = 0.0f;
        #pragma unroll
        for (int j = 0; j < KW; ++j) sum += rin[j];
        rout[0] = sum;
        #pragma unroll
        for (int i = 1; i < 16; ++i) {
            sum += rin[i + KW - 1] - rin[i - 1];
            rout[i] = sum;
        }
    }
    __syncthreads();

    // ---- Phase 3: vertical 11-tap sliding sums, vectorized 4 cols/thread ----
    // 32 column-groups x 8 row-groups of 4 rows = exactly 256 tasks.
    const int cg = tid & 31;                 // 4-col group (16B, coalesced)
    const int r0 = (tid >> 5) * 4;           // 4-row group
    const float inv = 1.0f / (float)(KW * KW);

    v4f sum = 0.0f;
    #pragma unroll
    for (int j = 0; j < KW; ++j)
        sum += *(const v4f*)&s_h[(r0 + j) * HS_STRIDE + cg * 4];

    float* ybase = yp + (size_t)(row0 + r0) * IMG_W + (col0 + cg * 4);
    __builtin_nontemporal_store(sum * inv, (v4f*)ybase);
    #pragma unroll
    for (int i = 1; i < 4; ++i) {
        sum += *(const v4f*)&s_h[(r0 + i + KW - 1) * HS_STRIDE + cg * 4]
             - *(const v4f*)&s_h[(r0 + i - 1)      * HS_STRIDE + cg * 4];
        __builtin_nontemporal_store(sum * inv, (v4f*)(ybase + (size_t)i * IMG_W));
    }
}

extern "C" void kernel_launch(void* const* d_in, const int* in_sizes, int n_in,
                              void* d_out, int out_size, void* d_ws, size_t ws_size,
                              hipStream_t stream) {
    (void)in_sizes; (void)n_in; (void)out_size; (void)d_ws; (void)ws_size;
    const float* x = (const float*)d_in[0];
    float*       y = (float*)d_out;
    dim3 grid(IMG_W / TILE_W, IMG_H / TILE_H, 8 * 64);  // (4, 16, 512)
    box_blur11_kernel<<<grid, NTHREADS, 0, stream>>>(x, y);
}